// DRRProjector_11828339933690
// MI455X (gfx1250) — compile-verified
//
#include <hip/hip_runtime.h>
#include <hip/hip_bf16.h>

// ---------------------------------------------------------------------------
// DRR projector for MI455X (gfx1250, wave32).
//   out[b,0,i,j] = 0.1 * sum_k mask * trilinear(vol_b, s_b + t_k * d_b(i,j))
// Volume: 4 x 128^3 fp32 (32 MB total -> L2 resident). Gather-bound out of L2.
// Ray generation (3x3 GEMM vs pixel coords) done with V_WMMA_F32_16X16X4_F32.
// ---------------------------------------------------------------------------

namespace {
constexpr int   VOLD   = 128;
constexpr int   VOLN   = VOLD * VOLD * VOLD;   // 2,097,152
constexpr int   NPIX   = 128 * 128;            // 16,384 detector pixels
constexpr int   NSTEPS = 226;
constexpr float ISOC   = 1000.0f;
constexpr float KDIAG  = 1.5f / 1500.0f;       // PIX/SDD
constexpr float KOFF   = -64.0f * (1.5f / 1500.0f); // -(DET/2)*PIX/SDD
}

typedef __attribute__((ext_vector_type(2))) float v2f;
typedef __attribute__((ext_vector_type(8))) float v8f;

__device__ __forceinline__ float trisample(const float* __restrict__ vol,
                                           float px, float py, float pz) {
  float fx = floorf(px), fy = floorf(py), fz = floorf(pz);
  int ix = (int)fx, iy = (int)fy, iz = (int)fz;
  float rx = px - fx, ry = py - fy, rz = pz - fz;
  float acc = 0.0f;
#pragma unroll
  for (int dx = 0; dx < 2; ++dx) {
    int   x  = ix + dx;
    float wx = dx ? rx : (1.0f - rx);
    bool  vx = (unsigned)x < (unsigned)VOLD;
    int   xc = min(max(x, 0), VOLD - 1);
#pragma unroll
    for (int dy = 0; dy < 2; ++dy) {
      int   y  = iy + dy;
      float wy = dy ? ry : (1.0f - ry);
      bool  vy = vx && ((unsigned)y < (unsigned)VOLD);
      int   yc = min(max(y, 0), VOLD - 1);
#pragma unroll
      for (int dz = 0; dz < 2; ++dz) {
        int   z  = iz + dz;
        float wz = dz ? rz : (1.0f - rz);
        bool  v  = vy && ((unsigned)z < (unsigned)VOLD);
        int   zc = min(max(z, 0), VOLD - 1);
        float s  = vol[(xc << 14) + (yc << 7) + zc]; // clamped, always in-buffer
        float w  = v ? (wx * wy * wz) : 0.0f;        // OOB corner contributes 0
        acc = fmaf(s, w, acc);
      }
    }
  }
  return acc;
}

__global__ __launch_bounds__(256) void drr_project_kernel(
    const float* __restrict__ vol4,   // (4,1,128,128,128) fp32
    const float* __restrict__ tp,     // (4,6) fp32
    float* __restrict__ out) {        // (4,1,128,128) fp32
  const int tid = blockIdx.x * 256 + threadIdx.x;
  const int b   = tid >> 14;               // batch
  const int p   = tid & (NPIX - 1);        // pixel, j fastest -> coalesced store
  const float U = (float)(p >> 7) + 0.5f;  // detector u (dim i)
  const float V = (float)(p & 127) + 0.5f; // detector v (dim j)
  const float* __restrict__ vol = vol4 + b * VOLN;

  // ---- per-batch geometry (broadcast loads, hit in cache) ----
  const float* tb = tp + b * 6;
  float thx = tb[0], thy = tb[1], thz = tb[2];
  float t0 = -tb[3], t1 = -tb[4], t2 = ISOC - tb[5];
  float cx = cosf(thx), sx = sinf(thx);
  float cy = cosf(thy), sy = sinf(thy);
  float cz = cosf(thz), sz = sinf(thz);
  // R = Rx @ Ry @ Rz
  float R00 = cy * cz,               R01 = -cy * sz,              R02 = sy;
  float R10 = cx * sz + sx * sy * cz, R11 = cx * cz - sx * sy * sz, R12 = -sx * cy;
  float R20 = sx * sz - cx * sy * cz, R21 = sx * cz + cx * sy * sz, R22 = cx * cy;
  // Rt rows (Rt[r][k] = R[k][r])
  float rt[3][3] = {{R00, R10, R20}, {R01, R11, R21}, {R02, R12, R22}};
  // ray_mat = Rt @ K_inv   (VOXINV = I, K_inv sparse upper-triangular)
  float rm[3][3];
#pragma unroll
  for (int r = 0; r < 3; ++r) {
    rm[r][0] = rt[r][0] * KDIAG;
    rm[r][1] = rt[r][1] * KDIAG;
    rm[r][2] = (rt[r][0] + rt[r][1]) * KOFF + rt[r][2];
  }
  float src[3];
#pragma unroll
  for (int r = 0; r < 3; ++r)
    src[r] = 64.0f - (rt[r][0] * t0 + rt[r][1] * t1 + rt[r][2] * t2);

  // ---- ray direction: dirs = ray_mat @ [U,V,1]  via WMMA f32 16x16x4 ----
  float dirx, diry, dirz;
#if __has_builtin(__builtin_amdgcn_wmma_f32_16x16x4_f32)
  {
    const int lane = threadIdx.x & 31;   // wave32
    const int m    = lane & 15;          // A-matrix row for this lane
    const bool lo  = lane < 16;
    // A 16x4 layout: lanes 0-15 hold K=0 (v0) / K=1 (v1); lanes 16-31 K=2/K=3.
    v2f A;
    A.x = (m < 3) ? (lo ? rm[m][0] : rm[m][2]) : 0.0f;
    A.y = (m < 3 && lo) ? rm[m][1] : 0.0f;
    // B 4x16 layout: v0 = row K0 (lanes 0-15) / row K2 (lanes 16-31); v1 = K1/K3.
    v2f B1;  B1.x = lo ? U : 1.0f;  B1.y = lo ? V : 0.0f;   // pixels base..base+15
    v8f C = {};
    v8f D1 = __builtin_amdgcn_wmma_f32_16x16x4_f32(
        false, A, false, B1, (short)0, C, false, false);
    // second half of the wave: columns = pixels base+16..base+31
    float U2 = __shfl_xor(U, 16, 32);
    float V2 = __shfl_xor(V, 16, 32);
    v2f B2;  B2.x = lo ? U2 : 1.0f;  B2.y = lo ? V2 : 0.0f;
    v8f D2 = __builtin_amdgcn_wmma_f32_16x16x4_f32(
        false, A, false, B2, (short)0, C, false, false);
    // D layout: VGPR r, lane n<16 -> D[M=r][N=n]; rows 0..2 are dirs components.
    float ex = __shfl_xor(D2[0], 16, 32);
    float ey = __shfl_xor(D2[1], 16, 32);
    float ez = __shfl_xor(D2[2], 16, 32);
    dirx = lo ? D1[0] : ex;
    diry = lo ? D1[1] : ey;
    dirz = lo ? D1[2] : ez;
  }
#else
  dirx = fmaf(rm[0][0], U, fmaf(rm[0][1], V, rm[0][2]));
  diry = fmaf(rm[1][0], U, fmaf(rm[1][1], V, rm[1][2]));
  dirz = fmaf(rm[2][0], U, fmaf(rm[2][1], V, rm[2][2]));
#endif

  // ---- normalize, AABB clip (matches reference safe_d semantics) ----
  float inv = rsqrtf(dirx * dirx + diry * diry + dirz * dirz);
  float ddx = dirx * inv, ddy = diry * inv, ddz = dirz * inv;
  float sdx = (fabsf(ddx) < 1e-8f) ? 1e-8f : ddx;
  float sdy = (fabsf(ddy) < 1e-8f) ? 1e-8f : ddy;
  float sdz = (fabsf(ddz) < 1e-8f) ? 1e-8f : ddz;
  float tax = (0.0f - src[0]) / sdx, tbx = (128.0f - src[0]) / sdx;
  float tay = (0.0f - src[1]) / sdy, tby = (128.0f - src[1]) / sdy;
  float taz = (0.0f - src[2]) / sdz, tbz = (128.0f - src[2]) / sdz;
  float tnear = fmaxf(fmaxf(fminf(tax, tbx), fminf(tay, tby)), fminf(taz, tbz));
  float tfar  = fminf(fminf(fmaxf(tax, tbx), fmaxf(tay, tby)), fmaxf(taz, tbz));
  float tmin  = fmaxf(tnear, 0.0f);

  // ---- march: incremental position, early-out at tfar ----
  float tcur = tmin + 0.5f;                    // STEP = 1
  float px = fmaf(tcur, ddx, src[0]);
  float py = fmaf(tcur, ddy, src[1]);
  float pz = fmaf(tcur, ddz, src[2]);
  float acc = 0.0f;
  for (int k = 0; k < NSTEPS; ++k) {
    if (tcur >= tfar) break;                   // mask = (t < tmax)
    acc += trisample(vol, px, py, pz);
    if ((k & 3) == 0) {                        // prefetch ~8 steps ahead
      float qx = fmaf(8.0f, ddx, px);
      float qy = fmaf(8.0f, ddy, py);
      float qz = fmaf(8.0f, ddz, pz);
      int qi = min(max((int)qx, 0), VOLD - 1);
      int qj = min(max((int)qy, 0), VOLD - 1);
      int qk = min(max((int)qz, 0), VOLD - 1);
      __builtin_prefetch(vol + ((qi << 14) + (qj << 7) + qk), 0, 0);
    }
    px += ddx; py += ddy; pz += ddz; tcur += 1.0f;
  }
  out[tid] = acc * 0.1f;                       // proj / 10
}

extern "C" void kernel_launch(void* const* d_in, const int* in_sizes, int n_in,
                              void* d_out, int out_size, void* d_ws, size_t ws_size,
                              hipStream_t stream) {
  (void)in_sizes; (void)n_in; (void)out_size; (void)d_ws; (void)ws_size;
  const float* vol = (const float*)d_in[0];   // (4,1,128,128,128) fp32
  const float* tp  = (const float*)d_in[1];   // (4,6) fp32
  float* out = (float*)d_out;                 // (4,1,128,128) fp32
  const int total  = 4 * NPIX;                // 65,536 threads
  const int block  = 256;                     // 8 wave32s
  drr_project_kernel<<<total / block, block, 0, stream>>>(vol, tp, out);
}